// GCN_Layer_39453569581648
// MI455X (gfx1250) — compile-verified
//
#include <hip/hip_runtime.h>

#define NEG_SLOPE 0.01f

typedef __attribute__((ext_vector_type(2))) float v2f;
typedef __attribute__((ext_vector_type(8))) float v8f;

// ---------------------------------------------------------------------------
// 1) init: degrees start at 1.0 (the appended self-loop), agg starts at 0
// ---------------------------------------------------------------------------
__global__ void gcn_init_deg(float* __restrict__ deg, int n2) {
    int i = blockIdx.x * blockDim.x + threadIdx.x;
    if (i < n2) deg[i] = 1.0f;
}

__global__ void gcn_zero_agg(float4* __restrict__ agg4, int total4) {
    int i = blockIdx.x * blockDim.x + threadIdx.x;
    if (i < total4) agg4[i] = make_float4(0.f, 0.f, 0.f, 0.f);
}

// ---------------------------------------------------------------------------
// 2) degree accumulation over edges
// ---------------------------------------------------------------------------
__global__ void gcn_degrees(const int* __restrict__ src, const int* __restrict__ dst,
                            float* __restrict__ deg_out, float* __restrict__ deg_in, int E) {
    int i = blockIdx.x * blockDim.x + threadIdx.x;
    if (i >= E) return;
    atomicAdd(&deg_out[src[i]], 1.0f);
    atomicAdd(&deg_in[dst[i]], 1.0f);
}

// ---------------------------------------------------------------------------
// 3) deg -> rsqrt(max(deg,1)) in place (covers both deg_out and deg_in: 2N)
// ---------------------------------------------------------------------------
__global__ void gcn_norms(float* __restrict__ deg, int n2) {
    int i = blockIdx.x * blockDim.x + threadIdx.x;
    if (i >= n2) return;
    deg[i] = rsqrtf(fmaxf(deg[i], 1.0f));
}

// ---------------------------------------------------------------------------
// 4) SpMM scatter: one wave (32 lanes) per edge; lane owns 4 consecutive
//    channels via a float4 load, then 4 f32 atomics into agg[dst].
//    x and agg are L2-resident (25.6 MB each << 192 MB L2).
// ---------------------------------------------------------------------------
__global__ __launch_bounds__(256) void gcn_scatter(const int* __restrict__ src,
                                                   const int* __restrict__ dst,
                                                   const float* __restrict__ x,
                                                   const float* __restrict__ norm_out,
                                                   float* __restrict__ agg, int E) {
    int t = blockIdx.x * blockDim.x + threadIdx.x;
    int e = t >> 5;
    int lane = t & 31;
    if (e >= E) return;
    int s = src[e];
    int d = dst[e];
    float ns = norm_out[s];
    float4 v = reinterpret_cast<const float4*>(x + (size_t)s * 128)[lane];
    float* ad = agg + (size_t)d * 128 + lane * 4;
    atomicAdd(ad + 0, v.x * ns);
    atomicAdd(ad + 1, v.y * ns);
    atomicAdd(ad + 2, v.z * ns);
    atomicAdd(ad + 3, v.w * ns);
}

// ---------------------------------------------------------------------------
// 5) fused self-loop add + dest-side normalization:
//    agg[i] = (agg[i] + x[i]*norm_out[i]) * norm_in[i]
// ---------------------------------------------------------------------------
__global__ void gcn_selfloop_scale(const float4* __restrict__ x4,
                                   const float* __restrict__ norm_out,
                                   const float* __restrict__ norm_in,
                                   float4* __restrict__ agg4, int total4) {
    int i = blockIdx.x * blockDim.x + threadIdx.x;
    if (i >= total4) return;
    int node = i >> 5;  // 32 float4 per node (128 ch)
    float no = norm_out[node];
    float ni = norm_in[node];
    float4 xv = x4[i];
    float4 av = agg4[i];
    av.x = (av.x + xv.x * no) * ni;
    av.y = (av.y + xv.y * no) * ni;
    av.z = (av.z + xv.z * no) * ni;
    av.w = (av.w + xv.w * no) * ni;
    agg4[i] = av;
}

// ---------------------------------------------------------------------------
// 6) Dense GEMM out = agg @ W + b, then LeakyReLU — full-precision f32 WMMA.
//    Block = 256 threads = 8 waves; block owns a 16-row strip, wave w owns
//    output columns [16w, 16w+16). K=128 -> 32 x V_WMMA_F32_16X16X4_F32.
//
//    A (16x4 f32) layout per ISA: lanes 0-15 VGPR0/1 = K0/K1, lanes 16-31 =
//    K2/K3 -> each lane loads a contiguous float2 at offset 2*(lane>>4).
//    B (4x16 f32): lanes 0-15 hold rows K0/K1 (VGPR0/1), lanes 16-31 K2/K3,
//    N = lane&15.
//    C/D (16x16 f32): VGPR r -> row r (lanes 0-15) / row r+8 (lanes 16-31).
// ---------------------------------------------------------------------------
__global__ __launch_bounds__(256) void gcn_gemm_wmma(const float* __restrict__ agg,
                                                     const float* __restrict__ W,
                                                     const float* __restrict__ bias,
                                                     float* __restrict__ out, int N) {
    int row0 = blockIdx.x * 16;

    if (row0 + 16 > N) {
        // Uniform-per-block scalar fallback (keeps WMMA path at EXEC==all-1s).
        for (int idx = threadIdx.x; idx < 16 * 128; idx += 256) {
            int r = idx >> 7;
            int c = idx & 127;
            int row = row0 + r;
            if (row < N) {
                float acc = bias[c];
                for (int k = 0; k < 128; ++k)
                    acc += agg[(size_t)row * 128 + k] * W[(size_t)k * 128 + c];
                out[(size_t)row * 128 + c] = acc >= 0.f ? acc : NEG_SLOPE * acc;
            }
        }
        return;
    }

    int wave = threadIdx.x >> 5;   // 0..7 -> which 16-col tile
    int lane = threadIdx.x & 31;
    int lm = lane & 15;
    int lh = lane >> 4;            // 0 or 1: selects K-pair within each step
    int col0 = wave * 16;

    const float* arow = agg + ((size_t)(row0 + lm)) * 128 + 2 * lh;   // A[M=lm][K base]
    const float* bcol = W + ((size_t)(2 * lh)) * 128 + col0 + lm;     // B[K base][N=lm]

    v8f acc = {};
    #pragma unroll
    for (int kk = 0; kk < 32; ++kk) {
        v2f a = *reinterpret_cast<const v2f*>(arow + 4 * kk);   // global_load_b64
        v2f bv;
        bv.x = bcol[(size_t)(4 * kk) * 128];
        bv.y = bcol[(size_t)(4 * kk + 1) * 128];
        acc = __builtin_amdgcn_wmma_f32_16x16x4_f32(
            /*neg_a=*/false, a, /*neg_b=*/false, bv,
            /*c_mod=*/(short)0, acc, /*reuse_a=*/false, /*reuse_b=*/false);
    }

    float bb = bias[col0 + lm];
    int rbase = row0 + 8 * lh;
    #pragma unroll
    for (int r = 0; r < 8; ++r) {
        float v = acc[r] + bb;
        v = v >= 0.f ? v : NEG_SLOPE * v;
        out[(size_t)(rbase + r) * 128 + col0 + lm] = v;
    }
}

// ---------------------------------------------------------------------------
// launch
// ---------------------------------------------------------------------------
extern "C" void kernel_launch(void* const* d_in, const int* in_sizes, int n_in,
                              void* d_out, int out_size, void* d_ws, size_t ws_size,
                              hipStream_t stream) {
    const int* src = (const int*)d_in[0];
    const int* dst = (const int*)d_in[1];
    const float* x = (const float*)d_in[2];
    const float* W = (const float*)d_in[3];
    const float* b = (const float*)d_in[4];
    float* out = (float*)d_out;

    const int E = in_sizes[0];
    const int N = in_sizes[2] / 128;

    // Workspace layout: [deg/norm_out: N][deg/norm_in: N][agg: N*128]
    float* norm_out = (float*)d_ws;
    float* norm_in  = norm_out + N;
    float* agg      = norm_in + N;

    const int BS = 256;

    gcn_init_deg<<<(2 * N + BS - 1) / BS, BS, 0, stream>>>(norm_out, 2 * N);

    int total4 = N * 32;  // N*128 floats as float4
    gcn_zero_agg<<<(total4 + BS - 1) / BS, BS, 0, stream>>>((float4*)agg, total4);

    gcn_degrees<<<(E + BS - 1) / BS, BS, 0, stream>>>(src, dst, norm_out, norm_in, E);

    gcn_norms<<<(2 * N + BS - 1) / BS, BS, 0, stream>>>(norm_out, 2 * N);

    long long scatter_threads = (long long)E * 32;
    gcn_scatter<<<(int)((scatter_threads + BS - 1) / BS), BS, 0, stream>>>(
        src, dst, x, norm_out, agg, E);

    gcn_selfloop_scale<<<(total4 + BS - 1) / BS, BS, 0, stream>>>(
        (const float4*)x, norm_out, norm_in, (float4*)agg, total4);

    int row_tiles = (N + 15) / 16;
    gcn_gemm_wmma<<<row_tiles, BS, 0, stream>>>(agg, W, b, out, N);
}